// LightGCN_17334488007154
// MI455X (gfx1250) — compile-verified
//
#include <hip/hip_runtime.h>
#include <hip/hip_bf16.h>

#define N_USERS 100000
#define N_ITEMS 50000
#define NNODES  150000
#define EMB     64
#define REG     1.0e-4f
#define BATCH   16384

typedef float v2f __attribute__((ext_vector_type(2)));
typedef float v8f __attribute__((ext_vector_type(8)));

// ---------------- init: acc = emb = concat(user_emb, item_emb); zero reduction slots
__global__ void lgcn_init(const float* __restrict__ ue, const float* __restrict__ ie,
                          float* __restrict__ acc, float* __restrict__ emb,
                          float* __restrict__ sums)
{
    long long idx = (long long)blockIdx.x * blockDim.x + threadIdx.x;
    if (idx == 0) { sums[0] = 0.0f; sums[1] = 0.0f; }
    const long long total = (long long)NNODES * EMB;
    if (idx < total) {
        const long long usz = (long long)N_USERS * EMB;
        float v = (idx < usz) ? ue[idx] : ie[idx - usz];
        acc[idx] = v;
        emb[idx] = v;
    }
}

// ---------------- zero a buffer
__global__ void lgcn_zero(float* __restrict__ p, long long n)
{
    long long idx = (long long)blockIdx.x * blockDim.x + threadIdx.x;
    if (idx < n) p[idx] = 0.0f;
}

// ---------------- SpMM scatter: out[row[e]] += vals[e] * emb[col[e]]
// 16 threads per edge, float4 per thread: a wave reads 2 full 256B rows coalesced.
__global__ void lgcn_spmm(const float* __restrict__ emb, float* __restrict__ out,
                          const float* __restrict__ vals, const int* __restrict__ row,
                          const int* __restrict__ col, int nnz)
{
    unsigned int gid = blockIdx.x * blockDim.x + threadIdx.x;
    int e = (int)(gid >> 4);
    int q = (int)(gid & 15u);
    if (e >= nnz) return;
    int r = row[e];
    int c = col[e];
    float v = vals[e];
    const float4* src = (const float4*)(emb + (long long)c * EMB) + q;
    float4 x = *src;
    float* dst = out + (long long)r * EMB + q * 4;
    atomicAdd(dst + 0, v * x.x);
    atomicAdd(dst + 1, v * x.y);
    atomicAdd(dst + 2, v * x.z);
    atomicAdd(dst + 3, v * x.w);
}

// ---------------- acc += e
__global__ void lgcn_accadd(float* __restrict__ acc, const float* __restrict__ e, long long n)
{
    long long idx = (long long)blockIdx.x * blockDim.x + threadIdx.x;
    if (idx < n) acc[idx] += e[idx];
}

// ---------------- BPR loss via V_WMMA_F32_16X16X4_F32
// One wave handles 16 batch elements. D = U x (P-N)^T accumulated over K=64 in
// chunks of 4; diag(D)[m] = y_ui[m] - y_uj[m], exact f32.
// A (16x4 f32, 2 VGPRs): lane l -> row M=l&15, K = {k0+2h, k0+2h+1}, h=l>>4.
// B (4x16 f32, 2 VGPRs): lane l -> col N=l&15, same K pattern -> identical addressing.
// C/D (16x16 f32, 8 VGPRs): D[m][m] on lane m (reg m) for m<8, lane m+16 (reg m-8) for m>=8.
__global__ void lgcn_loss(const float* __restrict__ acc,
                          const int* __restrict__ u, const int* __restrict__ i,
                          const int* __restrict__ j, float* __restrict__ sums)
{
    const int lane = (int)(threadIdx.x & 31u);
    const int wave = (int)(threadIdx.x >> 5);
    const int group = blockIdx.x * (blockDim.x >> 5) + wave;   // 16 elems per wave
    const int m    = lane & 15;
    const int half = lane >> 4;
    const int b    = group * 16 + m;

    const float* up = acc + (long long)u[b] * EMB;
    const float* pp = acc + (long long)i[b] * EMB;
    const float* np = acc + (long long)j[b] * EMB;

    v8f c = {0.f, 0.f, 0.f, 0.f, 0.f, 0.f, 0.f, 0.f};
    float l2 = 0.0f;

#pragma unroll
    for (int t = 0; t < 16; ++t) {
        const int k = 4 * t + 2 * half;
        float u0 = 0.25f * up[k],     u1 = 0.25f * up[k + 1];   // final = acc/4
        float p0 = 0.25f * pp[k],     p1 = 0.25f * pp[k + 1];
        float n0 = 0.25f * np[k],     n1 = 0.25f * np[k + 1];
        v2f a;  a.x = u0;       a.y = u1;
        v2f bb; bb.x = p0 - n0; bb.y = p1 - n1;
        c = __builtin_amdgcn_wmma_f32_16x16x4_f32(false, a, false, bb,
                                                  (short)0, c, false, false);
        l2 += u0*u0 + u1*u1 + p0*p0 + p1*p1 + n0*n0 + n1*n1;
    }

    // Extract diagonal of D held by this lane (if any).
    const int  dr      = lane & 7;
    const bool hasDiag = (lane < 8) || (lane >= 24);
    float cv = c[0];
    cv = (dr == 1) ? c[1] : cv;
    cv = (dr == 2) ? c[2] : cv;
    cv = (dr == 3) ? c[3] : cv;
    cv = (dr == 4) ? c[4] : cv;
    cv = (dr == 5) ? c[5] : cv;
    cv = (dr == 6) ? c[6] : cv;
    cv = (dr == 7) ? c[7] : cv;

    // -log_sigmoid(y) = softplus(-y), numerically stable
    float bpr = hasDiag ? (fmaxf(-cv, 0.0f) + log1pf(expf(-fabsf(cv)))) : 0.0f;

    // wave32 reduction
#pragma unroll
    for (int off = 16; off > 0; off >>= 1) {
        bpr += __shfl_xor(bpr, off, 32);
        l2  += __shfl_xor(l2,  off, 32);
    }
    if (lane == 0) {
        atomicAdd(&sums[0], bpr);
        atomicAdd(&sums[1], l2);
    }
}

// ---------------- finalize scalar
__global__ void lgcn_finalize(const float* __restrict__ sums, float* __restrict__ out)
{
    out[0] = sums[0] / (float)BATCH + REG * (sums[1] * 0.5f) / (float)BATCH;
}

extern "C" void kernel_launch(void* const* d_in, const int* in_sizes, int n_in,
                              void* d_out, int out_size, void* d_ws, size_t ws_size,
                              hipStream_t stream)
{
    const float* ue   = (const float*)d_in[0];
    const float* ie   = (const float*)d_in[1];
    const float* vals = (const float*)d_in[2];
    const int*   row  = (const int*)d_in[3];
    const int*   col  = (const int*)d_in[4];
    const int*   u    = (const int*)d_in[5];
    const int*   ii   = (const int*)d_in[6];
    const int*   jj   = (const int*)d_in[7];
    const int    nnz  = in_sizes[3];

    const long long TOT = (long long)NNODES * EMB;   // 9,600,000 floats
    float* ws   = (float*)d_ws;
    float* acc  = ws;
    float* bufA = ws + TOT;
    float* bufB = ws + 2 * TOT;
    float* sums = ws + 3 * TOT;

    const int TPB = 256;
    const int gTot = (int)((TOT + TPB - 1) / TPB);

    lgcn_init<<<gTot, TPB, 0, stream>>>(ue, ie, acc, bufA, sums);

    float* cur = bufA;
    float* nxt = bufB;
    const long long edgeThreads = (long long)nnz * 16;
    const int gEdge = (int)((edgeThreads + TPB - 1) / TPB);

    for (int layer = 0; layer < 3; ++layer) {
        lgcn_zero<<<gTot, TPB, 0, stream>>>(nxt, TOT);
        lgcn_spmm<<<gEdge, TPB, 0, stream>>>(cur, nxt, vals, row, col, nnz);
        lgcn_accadd<<<gTot, TPB, 0, stream>>>(acc, nxt, TOT);
        float* t = cur; cur = nxt; nxt = t;
    }

    // 1024 waves of 32, 8 waves per block -> 128 blocks, each wave does 16 elems
    lgcn_loss<<<BATCH / (16 * 8), TPB, 0, stream>>>(acc, u, ii, jj, sums);
    lgcn_finalize<<<1, 1, 0, stream>>>(sums, (float*)d_out);
}